// GaussianModel_39608188403926
// MI455X (gfx1250) — compile-verified
//
#include <hip/hip_runtime.h>

// MI455X (gfx1250) — bandwidth-bound streaming kernel.
// quat(4f) + scale(3f) -> cov(9f) per point; roofline ~22us at 23.3 TB/s.
// Strategy: NT-hinted wide loads, async global<->LDS DMA (ASYNCcnt path),
// LDS-staged fully-coalesced writeback, wave32.

typedef float v4f   __attribute__((ext_vector_type(4)));
typedef int   i32x4 __attribute__((vector_size(16)));   // matches builtin's int __vector(4)

#define TPB 256

// CPol immediate for gfx12+: TH[2:0]; TH=1 -> non-temporal (load & store).
// 512MB single-touch stream vs 192MB L2 -> keep it out of L2.
#define CPOL_NT 1

#if defined(__AMDGCN__) && __has_builtin(__builtin_amdgcn_global_load_async_to_lds_b32) && \
    __has_builtin(__builtin_amdgcn_global_store_async_from_lds_b128) && \
    __has_builtin(__builtin_amdgcn_global_store_async_from_lds_b32)
#define HAVE_ASYNC_LDS 1
#else
#define HAVE_ASYNC_LDS 0
#endif

// Exactly-typed, address-space-qualified casts (builtins take typed pointers:
// b32 <-> int*, b128 <-> int __vector(4)*; AS1 = global, AS3 = LDS).
#define AS_G_I32(p)   ((__attribute__((address_space(1))) int*)(p))
#define AS_L_I32(p)   ((__attribute__((address_space(3))) int*)(p))
#define AS_G_I32X4(p) ((__attribute__((address_space(1))) i32x4*)(p))
#define AS_L_I32X4(p) ((__attribute__((address_space(3))) i32x4*)(p))

__device__ __forceinline__ void wait_asynccnt0() {
#if __has_builtin(__builtin_amdgcn_s_wait_asynccnt)
    __builtin_amdgcn_s_wait_asynccnt(0);
#else
    asm volatile("s_wait_asynccnt 0x0" ::: "memory");
#endif
}

__global__ __launch_bounds__(TPB) void covar3d_kernel(
    const v4f*   __restrict__ rot,   // N x 4 (w,x,y,z)
    const float* __restrict__ scale, // N x 3
    float*       __restrict__ out,   // N x 9 (row-major 3x3)
    int N)
{
    __shared__ float               sS[TPB * 3];
    __shared__ __align__(16) float sO[TPB * 9];

    const int t    = threadIdx.x;
    const int base = blockIdx.x * TPB;
    const bool full = (base + TPB) <= N;

    // ---- stage scale: 768 contiguous dwords per block, fully coalesced ----
    if (full) {
#if HAVE_ASYNC_LDS
        // Async DMA global -> LDS (never transits VGPRs); overlaps with the
        // quaternion b128 load below. Tracked by ASYNCcnt.
        #pragma unroll
        for (int j = 0; j < 3; ++j) {
            __builtin_amdgcn_global_load_async_to_lds_b32(
                AS_G_I32(scale + base * 3 + j * TPB + t),
                AS_L_I32(&sS[j * TPB + t]),
                /*offset=*/0, /*cpol=*/CPOL_NT);
        }
#else
        #pragma unroll
        for (int j = 0; j < 3; ++j)
            sS[j * TPB + t] = __builtin_nontemporal_load(scale + base * 3 + j * TPB + t);
#endif
    } else {
        #pragma unroll
        for (int j = 0; j < 3; ++j) {
            int idx = base * 3 + j * TPB + t;
            sS[j * TPB + t] = (idx < 3 * N) ? scale[idx] : 0.0f;
        }
    }

    // ---- quaternion load proceeds while async copies are in flight ----
    const int i = base + t;
    v4f q;
    if (i < N) q = __builtin_nontemporal_load(rot + i);   // global_load_b128 th:NT
    else       q = (v4f){1.f, 0.f, 0.f, 0.f};

#if HAVE_ASYNC_LDS
    if (full) wait_asynccnt0();   // this wave's LDS data landed
#endif
    __syncthreads();              // all waves' staging visible

    // sS[k] mirrors scale[base*3 + k]; stride-3 dword read is conflict-free (gcd(3,64)=1)
    const float sx = sS[t * 3 + 0], sy = sS[t * 3 + 1], sz = sS[t * 3 + 2];

    const float inv = rsqrtf(q.x * q.x + q.y * q.y + q.z * q.z + q.w * q.w);
    const float w = q.x * inv, x = q.y * inv, y = q.z * inv, z = q.w * inv;

    const float xx = x * x, yy = y * y, zz = z * z;
    const float xy = x * y, xz = x * z, yz = y * z;
    const float wx = w * x, wy = w * y, wz = w * z;

    // M = R * diag(s)
    const float m00 = (1.f - 2.f * (yy + zz)) * sx;
    const float m01 = (2.f * (xy - wz))       * sy;
    const float m02 = (2.f * (xz + wy))       * sz;
    const float m10 = (2.f * (xy + wz))       * sx;
    const float m11 = (1.f - 2.f * (xx + zz)) * sy;
    const float m12 = (2.f * (yz - wx))       * sz;
    const float m20 = (2.f * (xz - wy))       * sx;
    const float m21 = (2.f * (yz + wx))       * sy;
    const float m22 = (1.f - 2.f * (xx + yy)) * sz;

    // cov = M M^T (symmetric: 6 unique dot products)
    const float c00 = m00 * m00 + m01 * m01 + m02 * m02;
    const float c01 = m00 * m10 + m01 * m11 + m02 * m12;
    const float c02 = m00 * m20 + m01 * m21 + m02 * m22;
    const float c11 = m10 * m10 + m11 * m11 + m12 * m12;
    const float c12 = m10 * m20 + m11 * m21 + m12 * m22;
    const float c22 = m20 * m20 + m21 * m21 + m22 * m22;

    // stride-9 dword write is conflict-free (gcd(9,64)=1)
    float* o = &sO[t * 9];
    o[0] = c00; o[1] = c01; o[2] = c02;
    o[3] = c01; o[4] = c11; o[5] = c12;
    o[6] = c02; o[7] = c12; o[8] = c22;

    __syncthreads();  // sO committed (dscnt drained) before LDS->global DMA

    // ---- coalesced writeback of 2304 contiguous dwords per block ----
    if (full) {
        float* ob = out + (size_t)base * 9;   // 9216B-aligned block base
#if HAVE_ASYNC_LDS
        // Async DMA LDS -> global: 2 x b128 + 1 x b32 per lane, all 16B aligned.
        #pragma unroll
        for (int j = 0; j < 2; ++j) {
            __builtin_amdgcn_global_store_async_from_lds_b128(
                AS_G_I32X4(ob + j * 1024 + 4 * t),
                AS_L_I32X4(&sO[j * 1024 + 4 * t]),
                /*offset=*/0, /*cpol=*/CPOL_NT);
        }
        __builtin_amdgcn_global_store_async_from_lds_b32(
            AS_G_I32(ob + 2048 + t),
            AS_L_I32(&sO[2048 + t]),
            /*offset=*/0, /*cpol=*/CPOL_NT);
        wait_asynccnt0();  // drain before wave teardown (S_ENDPGM would also wait-idle)
#else
        #pragma unroll
        for (int j = 0; j < 2; ++j) {
            v4f v = *reinterpret_cast<const v4f*>(&sO[j * 1024 + 4 * t]);
            __builtin_nontemporal_store(v, reinterpret_cast<v4f*>(ob + j * 1024 + 4 * t));
        }
        __builtin_nontemporal_store(sO[2048 + t], ob + 2048 + t);
#endif
    } else {
        #pragma unroll
        for (int j = 0; j < 9; ++j) {
            int idx = base * 9 + j * TPB + t;
            if (idx < 9 * N) out[idx] = sO[j * TPB + t];
        }
    }
}

extern "C" void kernel_launch(void* const* d_in, const int* in_sizes, int n_in,
                              void* d_out, int out_size, void* d_ws, size_t ws_size,
                              hipStream_t stream) {
    const v4f*   rot   = (const v4f*)d_in[0];   // float32 N x 4
    const float* scale = (const float*)d_in[1]; // float32 N x 3
    float*       out   = (float*)d_out;         // float32 N x 3 x 3

    const int N = in_sizes[0] / 4;              // 8,000,000
    const int blocks = (N + TPB - 1) / TPB;     // 31,250 (exact)

    covar3d_kernel<<<blocks, TPB, 0, stream>>>(rot, scale, out, N);
}